// ExternalKnowledge_43851616092292
// MI455X (gfx1250) — compile-verified
//
#include <hip/hip_runtime.h>

#define B_ 32
#define M_ 8192
#define D_ 128
#define HOPS_ 3
#define NEG_BIG (-1.0e10f)

#define CHUNKS 16
#define ROWS_PER_CHUNK (M_ / CHUNKS) /* 512 */

typedef __attribute__((ext_vector_type(2))) float v2f;
typedef __attribute__((ext_vector_type(8))) float v8f;

// ---------------- init: u = query_vector ----------------
__global__ void k_init_u(const float* __restrict__ q, float* __restrict__ u) {
    int i = blockIdx.x * blockDim.x + threadIdx.x;
    if (i < B_ * D_) u[i] = q[i];
}

// ---------------- logits via V_WMMA_F32_16X16X4_F32 ----------------
// grid (16, B), block 256 (8 waves). Each wave: 4 tiles of 16 rows.
// A tile = 16 rows x 4 K of m_story[hop]; B = u chunk broadcast to all 16 cols.
// Accumulate 32 WMMAs over D=128; every column of D holds the 16 row-dots.
__global__ void k_logits(const float* __restrict__ mA,   // m_story[hop] [B,M,D]
                         const float* __restrict__ u,    // [B,D] (ws)
                         const float* __restrict__ gp,
                         const float* __restrict__ la,
                         const int*   __restrict__ rm,
                         const int*   __restrict__ mm,
                         float* __restrict__ logits)     // [B,M]
{
    __shared__ __align__(16) float su[D_];
    const int b   = blockIdx.y;
    const int tid = threadIdx.x;
    if (tid < D_) su[tid] = u[b * D_ + tid];
    __syncthreads();

    const int lane   = tid & 31;
    const int wave   = tid >> 5;
    const int kk_off = (lane >> 4) << 1;   // lanes 0-15 -> K=0,1 ; lanes 16-31 -> K=2,3
    const int subm   = lane & 15;
    const size_t bbase = (size_t)b * M_ * D_;

    const int tile0 = (blockIdx.x * 8 + wave) * 4;   // 512 tiles per batch total
    for (int t = 0; t < 4; ++t) {
        const int m0 = (tile0 + t) * 16;
        const float* arow = mA + bbase + (size_t)(m0 + subm) * D_;
        v8f acc = {};
        #pragma unroll 4
        for (int k = 0; k < D_; k += 4) {
            const int kk = k + kk_off;
            // dead stream after this hop -> non-temporal, keep L2 for m_C reuse
            v2f a = __builtin_nontemporal_load((const v2f*)(arow + kk));
            v2f bf = *(const v2f*)(&su[kk]);   // broadcast u chunk to all 16 cols
            acc = __builtin_amdgcn_wmma_f32_16x16x4_f32(
                false, a, false, bf, (short)0, acc, false, false);
        }
        // D layout: lanes 0-15 hold rows m0+0..7 in acc[0..7]; lanes 16-31 rows m0+8..15
        if (subm < 8) {
            const int row = m0 + subm + ((lane >> 4) << 3);
            const int idx = b * M_ + row;
            float v = acc[subm];
            v *= gp[idx] * la[idx];                      // gate applied post-dot (scalar/row)
            if (rm[idx] == 0) v = -__builtin_inff();     // reason mask
            v += (1.0f - (float)mm[idx]) * NEG_BIG;      // memory mask
            logits[idx] = v;
        }
    }
}

// ---------------- softmax over M per batch ----------------
__global__ void k_softmax(const float* __restrict__ logits, float* __restrict__ prob) {
    __shared__ float red[256];
    const int b = blockIdx.x, tid = threadIdx.x;
    const float* L = logits + b * M_;
    float* P = prob + b * M_;

    float mx = -__builtin_inff();
    for (int m = tid; m < M_; m += 256) mx = fmaxf(mx, L[m]);
    red[tid] = mx; __syncthreads();
    for (int s = 128; s > 0; s >>= 1) {
        if (tid < s) red[tid] = fmaxf(red[tid], red[tid + s]);
        __syncthreads();
    }
    const float gmax = red[0];
    __syncthreads();

    float sum = 0.0f;
    for (int m = tid; m < M_; m += 256) {
        float e = __expf(L[m] - gmax);
        P[m] = e;
        sum += e;
    }
    red[tid] = sum; __syncthreads();
    for (int s = 128; s > 0; s >>= 1) {
        if (tid < s) red[tid] += red[tid + s];
        __syncthreads();
    }
    const float inv = 1.0f / red[0];
    __syncthreads();
    for (int m = tid; m < M_; m += 256) P[m] *= inv;
}

// ---------------- o_k partials: coalesced stream of m_C ----------------
// grid (CHUNKS, B), block 128 (lane == d). Regular (temporal) loads so m_C
// stays in 192MB L2 for next hop's logits pass.
__global__ void k_ok_partial(const float* __restrict__ mC,    // m_story[hop+1]
                             const float* __restrict__ prob,
                             const float* __restrict__ gp,
                             const float* __restrict__ la,
                             float* __restrict__ partial)     // [B, CHUNKS, D]
{
    __shared__ float w[ROWS_PER_CHUNK];
    const int c = blockIdx.x, b = blockIdx.y, tid = threadIdx.x;
    const int mbase = c * ROWS_PER_CHUNK;

    for (int i = tid; i < ROWS_PER_CHUNK; i += D_) {
        const int idx = b * M_ + mbase + i;
        w[i] = prob[idx] * gp[idx] * la[idx];
    }
    __syncthreads();

    const float* src = mC + (size_t)b * M_ * D_ + (size_t)mbase * D_ + tid;
    float acc = 0.0f;
    #pragma unroll 4
    for (int i = 0; i < ROWS_PER_CHUNK; ++i)
        acc = fmaf(w[i], src[(size_t)i * D_], acc);

    partial[((b * CHUNKS) + c) * D_ + tid] = acc;
}

// ---------------- reduce partials, update u ----------------
__global__ void k_ok_reduce(const float* __restrict__ partial,
                            float* __restrict__ u,       // ws working copy
                            float* __restrict__ u_out)   // d_out u region
{
    const int b = blockIdx.x, d = threadIdx.x;
    float s = 0.0f;
    #pragma unroll
    for (int c = 0; c < CHUNKS; ++c) s += partial[((b * CHUNKS) + c) * D_ + d];
    const float nu = u[b * D_ + d] + s;
    u[b * D_ + d] = nu;
    u_out[b * D_ + d] = nu;
}

extern "C" void kernel_launch(void* const* d_in, const int* in_sizes, int n_in,
                              void* d_out, int out_size, void* d_ws, size_t ws_size,
                              hipStream_t stream) {
    const float* q  = (const float*)d_in[0];   // query_vector [B,D]
    const float* ms = (const float*)d_in[1];   // m_story [HOPS+1,B,M,D]
    const float* gp = (const float*)d_in[2];   // global_pointer [B,M]
    const float* la = (const float*)d_in[3];   // local_attention [B,M]
    const int*   rm = (const int*)d_in[4];     // reason_mask [B,M]
    const int*   mm = (const int*)d_in[5];     // memory_mask [B,M]

    float* out    = (float*)d_out;
    float* prob   = out;                       // [B,M]
    float* logits = out + (size_t)B_ * M_;     // [B,M]
    float* u_out  = out + (size_t)2 * B_ * M_; // [B,D]

    float* ws      = (float*)d_ws;
    float* u_cur   = ws;                       // B*D floats
    float* partial = ws + B_ * D_;             // B*CHUNKS*D floats

    k_init_u<<<(B_ * D_ + 255) / 256, 256, 0, stream>>>(q, u_cur);

    for (int hop = 0; hop < HOPS_; ++hop) {
        const float* mA = ms + (size_t)hop * B_ * M_ * D_;
        const float* mC = ms + (size_t)(hop + 1) * B_ * M_ * D_;
        k_logits   <<<dim3(16, B_),     256, 0, stream>>>(mA, u_cur, gp, la, rm, mm, logits);
        k_softmax  <<<B_,               256, 0, stream>>>(logits, prob);
        k_ok_partial<<<dim3(CHUNKS, B_), D_, 0, stream>>>(mC, prob, gp, la, partial);
        k_ok_reduce<<<B_,                D_, 0, stream>>>(partial, u_cur, u_out);
    }
}